// RL_AR_Tree_8873402433917
// MI455X (gfx1250) — compile-verified
//
#include <hip/hip_runtime.h>
#include <hip/hip_bf16.h>
#include <math.h>

// ---------------------------------------------------------------------------
// RL_AR_Tree on MI455X (gfx1250, wave32, WMMA).
// All GEMMs use V_WMMA_F32_16X16X4_F32 (exact fp32, matches reference dtype).
// Batch dim B=16 == WMMA M=16.  Sequential recurrences synchronized by kernel
// launch boundaries (graph-capture safe).  LDS operand staging uses
// GLOBAL_LOAD_ASYNC_TO_LDS_B128 (ASYNCcnt) — direct global->LDS, no VGPR trip.
// ---------------------------------------------------------------------------

typedef __attribute__((ext_vector_type(2))) float v2f;
typedef __attribute__((ext_vector_type(8))) float v8f;

#define Bb   16
#define Ll   64
#define Dd   1024
#define Hh   1024
#define G4   4096      // 4*H
#define G6   6144      // 6*H
#define K3   3072      // 3*H
#define AS1  1032      // padded LDS row stride for 1024-wide A (bank-conflict avoid)
#define AS2  3080      // padded LDS row stride for 3072-wide A

__device__ inline float sigmoidf_(float x) { return 1.0f / (1.0f + expf(-x)); }

__device__ inline v2f ld2(const float* p) {      // 8B-aligned pair load
  float2 t = *reinterpret_cast<const float2*>(p);
  v2f r; r.x = t.x; r.y = t.y; return r;
}

// D = A(16x4) * B(4x16) + C(16x16), fp32 exact.
__device__ inline v8f wmma4(v2f a, v2f b, v8f c) {
  return __builtin_amdgcn_wmma_f32_16x16x4_f32(
      /*neg_a=*/false, a, /*neg_b=*/false, b,
      /*c_mod=*/(short)0, c, /*reuse_a=*/false, /*reuse_b=*/false);
}

// Async global -> LDS copy of 16B (per active lane).  LDS flat address [31:0]
// is the LDS byte offset (aperture rule), which is what VDST must hold.
__device__ inline void async_ld16(float* ldsDst, const float* gsrc) {
  unsigned ldsOff = (unsigned)(unsigned long long)(void*)ldsDst;
  asm volatile("global_load_async_to_lds_b128 %0, %1, off"
               :
               : "v"(ldsOff), "v"(gsrc)
               : "memory");
}
__device__ inline void wait_async0() {
  asm volatile("s_wait_asynccnt 0x0" ::: "memory");
}

// ---------------------------------------------------------------------------
// K1: X[r, col] = emb[r, :] . w_ih[col, :] + b_ih[col] + b_hh[col]
//     r in [0,1024) = b*64+l, col in [0,4096).  One wave per 16x16 tile.
// ---------------------------------------------------------------------------
__global__ __launch_bounds__(128) void gemm_xw(
    const float* __restrict__ emb, const float* __restrict__ w_ih,
    const float* __restrict__ b_ih, const float* __restrict__ b_hh,
    float* __restrict__ X) {
  int wave = threadIdx.x >> 5, lane = threadIdx.x & 31;
  int half = lane >> 4, lm = lane & 15;
  int tile = blockIdx.x * 4 + wave;          // 16384 tiles
  int mt = tile >> 8, nt = tile & 255;
  int m0 = mt * 16, n0 = nt * 16;
  const float* pa = emb  + (size_t)(m0 + lm) * Dd + half * 2;  // A[m][k]
  const float* pb = w_ih + (size_t)(n0 + lm) * Dd + half * 2;  // B[k][n]=w_ih[n][k]
  v8f acc = {};
#pragma unroll 4
  for (int k = 0; k < Dd; k += 4)
    acc = wmma4(ld2(pa + k), ld2(pb + k), acc);
  float bias = b_ih[n0 + lm] + b_hh[n0 + lm];
#pragma unroll
  for (int r = 0; r < 8; ++r)
    X[(size_t)(m0 + r + half * 8) * G4 + n0 + lm] = acc[r] + bias;
}

// ---------------------------------------------------------------------------
// K2: rank scores: sc[r] = relu(emb[r,:] @ w1.T) @ w2.T   (128 hidden units)
// ---------------------------------------------------------------------------
__global__ __launch_bounds__(128) void rank_scores(
    const float* __restrict__ emb, const float* __restrict__ w1,
    const float* __restrict__ w2, float* __restrict__ sc) {
  __shared__ float red[128];
  int row = blockIdx.x, u = threadIdx.x;
  const float* e = emb + (size_t)row * Dd;
  const float* w = w1 + (size_t)u * Dd;
  float acc = 0.f;
  for (int k = 0; k < Dd; k += 4) {
    float4 ev = *reinterpret_cast<const float4*>(e + k);
    float4 wv = *reinterpret_cast<const float4*>(w + k);
    acc += ev.x * wv.x + ev.y * wv.y + ev.z * wv.z + ev.w * wv.w;
  }
  red[u] = fmaxf(acc, 0.f) * w2[u];
  __syncthreads();
  for (int s = 64; s > 0; s >>= 1) {
    if (u < s) red[u] += red[u + s];
    __syncthreads();
  }
  if (u == 0) sc[row] = red[0];
}

// ---------------------------------------------------------------------------
// K3: greedy tree build (post-order) + REINFORCE log-prob sum.  1 thread/sent.
// node record kinds: 0=None(zeros), 1=leaf(idx=time pos), 2=internal(idx=step)
// ---------------------------------------------------------------------------
__global__ void build_trees(const float* __restrict__ scores,
                            const int* __restrict__ length,
                            int* __restrict__ npos, int* __restrict__ lkind,
                            int* __restrict__ lidx, int* __restrict__ rkind,
                            int* __restrict__ ridx, int* __restrict__ cnt,
                            float* __restrict__ lp_out) {
  if (threadIdx.x != 0) return;
  int b = blockIdx.x;
  const float* sc = scores + b * Ll;
  int len = length[b];
  int fs[Ll + 2], fe[Ll + 2], fp[Ll + 2], fst[Ll + 2], flk[Ll + 2], fli[Ll + 2];
  int sp = 0;
  fs[0] = 0; fe[0] = len; fst[0] = 0;
  int retk = 0, reti = 0, n = 0;
  float lp = 0.f;
  while (sp >= 0) {
    int s = fs[sp], e = fe[sp];
    if (e <= s)     { retk = 0; reti = 0; --sp; continue; }
    if (e == s + 1) { retk = 1; reti = s; --sp; continue; }
    int st = fst[sp];
    if (st == 0) {
      int am = s; float mv = sc[s];
      for (int i = s + 1; i < e; ++i) if (sc[i] > mv) { mv = sc[i]; am = i; }
      fp[sp] = am; fst[sp] = 1;
      ++sp; fs[sp] = s; fe[sp] = am; fst[sp] = 0;           // left
    } else if (st == 1) {
      flk[sp] = retk; fli[sp] = reti; fst[sp] = 2;
      int p = fp[sp];
      ++sp; fs[sp] = p + 1; fe[sp] = e; fst[sp] = 0;        // right
    } else {
      int p = fp[sp];
      int g = b * Ll + n;
      npos[g] = p; lkind[g] = flk[sp]; lidx[g] = fli[sp];
      rkind[g] = retk; ridx[g] = reti;
      float mv = sc[s];
      for (int i = s + 1; i < e; ++i) mv = fmaxf(mv, sc[i]);
      float ssum = 0.f;
      for (int i = s; i < e; ++i) ssum += expf(sc[i] - mv);
      lp += (float)(e - s) * (sc[p] - mv - logf(ssum));
      retk = 2; reti = n; ++n; --sp;
    }
  }
  cnt[b] = n;
  lp_out[b] = lp;
}

// ---------------------------------------------------------------------------
// K4 (x64): one LSTM step.  Block = 16-wide hidden tile; 4 waves = 4 gates.
// h_prev async-staged in LDS (padded stride), WMMA over K=1024, fused update.
// ---------------------------------------------------------------------------
__global__ __launch_bounds__(128) void lstm_step(
    const float* __restrict__ X, const float* __restrict__ w_hh,
    const float* __restrict__ h_prev, float* __restrict__ h_out,
    float* __restrict__ c_buf, float* __restrict__ hs, float* __restrict__ cs,
    int t) {
  extern __shared__ float lds[];
  float* A = lds;                 // 16 * AS1
  float* G = lds + 16 * AS1;      // 4 * 256 gate tiles
  int tid = threadIdx.x;
  for (int i = tid; i < 16 * 256; i += 128) {   // h_prev (16x1024) -> LDS
    int m = i >> 8, q = i & 255;
    async_ld16(A + m * AS1 + q * 4, h_prev + m * Hh + q * 4);
  }
  wait_async0();
  __syncthreads();
  int wave = tid >> 5, lane = tid & 31, half = lane >> 4, lm = lane & 15;
  int j0 = blockIdx.x * 16;
  int col0 = wave * Hh + j0;                    // gate-column base
  const float* pb = w_hh + (size_t)(col0 + lm) * Hh + half * 2;
  const float* pa = A + lm * AS1 + half * 2;
  v8f acc = {};
#pragma unroll 4
  for (int k = 0; k < Hh; k += 4)
    acc = wmma4(ld2(pa + k), ld2(pb + k), acc);
#pragma unroll
  for (int r = 0; r < 8; ++r) {
    int m = r + half * 8;
    G[wave * 256 + m * 16 + lm] =
        acc[r] + X[((size_t)m * Ll + t) * G4 + col0 + lm];
  }
  __syncthreads();
  for (int e = tid; e < 256; e += 128) {
    int m = e >> 4, j = e & 15, idx = m * Hh + j0 + j;
    float gi = G[e], gf = G[256 + e], gg = G[512 + e], go = G[768 + e];
    float cp = c_buf[idx];
    float cn = sigmoidf_(gf) * cp + sigmoidf_(gi) * tanhf(gg);
    float hn = sigmoidf_(go) * tanhf(cn);
    c_buf[idx] = cn; h_out[idx] = hn;
    size_t hix = ((size_t)m * Ll + t) * Hh + j0 + j;
    hs[hix] = hn; cs[hix] = cn;
  }
}

// ---------------------------------------------------------------------------
// K5a (x63): gather children for tree step k into Au=[hl|hr|hm], Cu=[cl|cr|cm]
// ---------------------------------------------------------------------------
__global__ __launch_bounds__(1024) void tree_gather(
    const float* __restrict__ hs, const float* __restrict__ cs,
    const float* __restrict__ h_node, const float* __restrict__ c_node,
    const int* __restrict__ npos, const int* __restrict__ lkind,
    const int* __restrict__ lidx, const int* __restrict__ rkind,
    const int* __restrict__ ridx, float* __restrict__ Au,
    float* __restrict__ Cu, int k) {
  int b = blockIdx.x / 3, seg = blockIdx.x % 3, j = threadIdx.x;
  int g = b * Ll + k;
  float h, c;
  if (seg == 2) {
    int p = npos[g];
    size_t ix = ((size_t)b * Ll + p) * Hh + j;
    h = hs[ix]; c = cs[ix];
  } else {
    int kind = (seg == 0) ? lkind[g] : rkind[g];
    int idx  = (seg == 0) ? lidx[g]  : ridx[g];
    if (kind == 0) { h = 0.f; c = 0.f; }
    else if (kind == 1) {
      size_t ix = ((size_t)b * Ll + idx) * Hh + j;
      h = hs[ix]; c = cs[ix];
    } else {
      size_t ix = ((size_t)b * Ll + idx) * Hh + j;
      h = h_node[ix]; c = c_node[ix];
    }
  }
  size_t o = (size_t)b * K3 + seg * Hh + j;
  Au[o] = h; Cu[o] = c;
}

// ---------------------------------------------------------------------------
// K5b (x63): TreeLSTM node step.  Block = 16-wide hidden tile; 6 waves=6 gates.
// A (16x3072) async-staged in LDS, WMMA over K=3072, fused update.
// ---------------------------------------------------------------------------
__global__ __launch_bounds__(192) void tree_step(
    const float* __restrict__ Au, const float* __restrict__ Cu,
    const float* __restrict__ tl_w, const float* __restrict__ tl_b,
    float* __restrict__ h_node, float* __restrict__ c_node, int k) {
  extern __shared__ float lds[];
  float* A = lds;                 // 16 * AS2
  float* G = lds + 16 * AS2;      // 6 * 256
  int tid = threadIdx.x;
  for (int i = tid; i < 16 * 768; i += 192) {   // Au (16x3072) -> LDS
    int m = i / 768, q = i % 768;
    async_ld16(A + m * AS2 + q * 4, Au + (size_t)m * K3 + q * 4);
  }
  wait_async0();
  __syncthreads();
  int wave = tid >> 5, lane = tid & 31, half = lane >> 4, lm = lane & 15;
  int j0 = blockIdx.x * 16;
  int col0 = wave * Hh + j0;                    // gate-column base (6 gates)
  const float* pb = tl_w + (size_t)(col0 + lm) * K3 + half * 2;
  const float* pa = A + lm * AS2 + half * 2;
  v8f acc = {};
#pragma unroll 4
  for (int kk = 0; kk < K3; kk += 4)
    acc = wmma4(ld2(pa + kk), ld2(pb + kk), acc);
#pragma unroll
  for (int r = 0; r < 8; ++r) {
    int m = r + half * 8;
    G[wave * 256 + m * 16 + lm] = acc[r] + tl_b[col0 + lm];
  }
  __syncthreads();
  for (int e = tid; e < 256; e += 192) {
    int m = e >> 4, j = e & 15;
    size_t co = (size_t)m * K3 + j0 + j;
    float cl = Cu[co], cr = Cu[co + Hh], cm = Cu[co + 2 * Hh];
    float gi = G[e], gfl = G[256 + e], gfr = G[512 + e];
    float gfm = G[768 + e], gu = G[1024 + e], go = G[1280 + e];
    float cn = sigmoidf_(gfl) * cl + sigmoidf_(gfr) * cr +
               sigmoidf_(gfm) * cm + sigmoidf_(gi) * tanhf(gu);
    float hn = sigmoidf_(go) * tanhf(cn);
    size_t ox = ((size_t)m * Ll + k) * Hh + j0 + j;
    h_node[ox] = hn; c_node[ox] = cn;
  }
}

// ---------------------------------------------------------------------------
// K6: root states -> d_out
// ---------------------------------------------------------------------------
__global__ __launch_bounds__(1024) void finalize_k(
    const float* __restrict__ h_node, const float* __restrict__ c_node,
    const int* __restrict__ cnt, float* __restrict__ out) {
  int b = blockIdx.x, j = threadIdx.x;
  int root = cnt[b] - 1;
  size_t ix = ((size_t)b * Ll + root) * Hh + j;
  out[(size_t)b * Hh + j] = h_node[ix];
  out[(size_t)Bb * Hh + (size_t)b * Hh + j] = c_node[ix];
}

// ---------------------------------------------------------------------------
extern "C" void kernel_launch(void* const* d_in, const int* in_sizes, int n_in,
                              void* d_out, int out_size, void* d_ws,
                              size_t ws_size, hipStream_t stream) {
  const float* emb     = (const float*)d_in[0];   // (16,64,1024)
  // d_in[1] sentence_word: unused by the computation
  const int*   length  = (const int*)d_in[2];     // (16,)
  const float* w_ih    = (const float*)d_in[3];   // (4096,1024)
  const float* w_hh    = (const float*)d_in[4];   // (4096,1024)
  const float* b_ih    = (const float*)d_in[5];   // (4096,)
  const float* b_hh    = (const float*)d_in[6];   // (4096,)
  const float* tl_w    = (const float*)d_in[7];   // (6144,3072)
  const float* tl_b    = (const float*)d_in[8];   // (6144,)
  const float* rank_w1 = (const float*)d_in[9];   // (128,1024)
  const float* rank_w2 = (const float*)d_in[10];  // (1,128)
  float* out = (float*)d_out;                     // h(16K) | c(16K) | lps(16)

  // ---- workspace carve-up (256B aligned; ~34.3 MB total) ----
  char* w = (char*)d_ws;
  auto alloc = [&](size_t nbytes) -> char* {
    char* p = w; w += (nbytes + 255) & ~(size_t)255; return p;
  };
  float* X      = (float*)alloc((size_t)Bb * Ll * G4 * 4);  // 16 MB
  float* hs     = (float*)alloc((size_t)Bb * Ll * Hh * 4);  // 4 MB
  float* cs     = (float*)alloc((size_t)Bb * Ll * Hh * 4);  // 4 MB
  float* sc     = (float*)alloc((size_t)Bb * Ll * 4);
  float* hbufA  = (float*)alloc((size_t)Bb * Hh * 4);       // |
  float* hbufB  = (float*)alloc((size_t)Bb * Hh * 4);       // | contiguous,
  float* cbuf   = (float*)alloc((size_t)Bb * Hh * 4);       // | one memset
  float* h_node = (float*)alloc((size_t)Bb * Ll * Hh * 4);  // 4 MB
  float* c_node = (float*)alloc((size_t)Bb * Ll * Hh * 4);  // 4 MB
  float* Au     = (float*)alloc((size_t)Bb * K3 * 4);
  float* Cu     = (float*)alloc((size_t)Bb * K3 * 4);
  int*   irec   = (int*)alloc((size_t)(5 * Bb * Ll + Bb) * 4);
  int* npos  = irec;
  int* lkind = irec + 1 * Bb * Ll;
  int* lidx  = irec + 2 * Bb * Ll;
  int* rkind = irec + 3 * Bb * Ll;
  int* ridx  = irec + 4 * Bb * Ll;
  int* cnt   = irec + 5 * Bb * Ll;

  // zero-init: h0/c0 state + node records (capture-safe stream memsets)
  hipMemsetAsync(hbufA, 0, (size_t)3 * Bb * Hh * 4, stream);   // hbufA..cbuf
  hipMemsetAsync(irec, 0, (size_t)(5 * Bb * Ll + Bb) * 4, stream);

  // Phase 1: input-gate GEMM + rank scores + tree build
  gemm_xw<<<4096, 128, 0, stream>>>(emb, w_ih, b_ih, b_hh, X);
  rank_scores<<<Bb * Ll, 128, 0, stream>>>(emb, rank_w1, rank_w2, sc);
  build_trees<<<Bb, 32, 0, stream>>>(sc, length, npos, lkind, lidx, rkind,
                                     ridx, cnt, out + 2 * Bb * Hh);

  // Phase 2: 64 sequential LSTM steps (double-buffered h)
  float* hb[2] = {hbufA, hbufB};
  const size_t lds1 = (size_t)(16 * AS1 + 4 * 256) * 4;
  for (int t = 0; t < Ll; ++t)
    lstm_step<<<Hh / 16, 128, lds1, stream>>>(X, w_hh, hb[t & 1],
                                              hb[(t + 1) & 1], cbuf, hs, cs, t);

  // Phase 3: 63 post-order TreeLSTM node steps (batch-16 across sentences)
  const size_t lds2 = (size_t)(16 * AS2 + 6 * 256) * 4;
  for (int k = 0; k < Ll - 1; ++k) {
    tree_gather<<<Bb * 3, 1024, 0, stream>>>(hs, cs, h_node, c_node, npos,
                                             lkind, lidx, rkind, ridx, Au, Cu, k);
    tree_step<<<Hh / 16, 192, lds2, stream>>>(Au, Cu, tl_w, tl_b, h_node,
                                              c_node, k);
  }

  // Phase 4: roots -> output
  finalize_k<<<Bb, 1024, 0, stream>>>(h_node, c_node, cnt, out);
}